// EOS_73091753443590
// MI455X (gfx1250) — compile-verified
//
#include <hip/hip_runtime.h>

#ifndef __has_builtin
#define __has_builtin(x) 0
#endif

// Problem constants (match reference setup_inputs)
#define B_ 4
#define N_ 4096
#define D_ 512
#define K_ 8
#define DT 256      // d-values per block (2 blocks cover d=512)
#define TMAX 256    // max chunk length supported by static LDS

// LDS f-buffer uses the SAME k-stride as global f (512 floats = 2048 B) so a
// single imm offset (added by HW to BOTH the LDS and global address) addresses
// both sides; one VGPR address pair serves all 8 async copies per step.
#define FSTRIDE 512                 // floats per k in LDS buffer
#define FBUFSZ  (K_ * FSTRIDE)      // floats per double-buffer half (== K_*D_)

#define GAS __attribute__((address_space(1)))
#define LAS __attribute__((address_space(3)))

#if __has_builtin(__builtin_amdgcn_global_load_async_to_lds_b32)
#define HAVE_ASYNC_LDS 1
#endif

// One f32 global -> LDS copy at compile-time byte offset OFF (applied to both
// addresses by the hardware). Async DMA on gfx1250 (ASYNCcnt-tracked).
template <int OFF>
__device__ __forceinline__ void cp_off(float* lds_base, const float* g_base) {
#ifdef HAVE_ASYNC_LDS
  __builtin_amdgcn_global_load_async_to_lds_b32(
      (GAS int*)(unsigned long long)(const void*)g_base,
      (LAS int*)lds_base, OFF, 0);
#else
  *(float*)((char*)lds_base + OFF) =
      __builtin_nontemporal_load((const float*)((const char*)g_base + OFF));
#endif
}

// Stage one timestep: 8 k-slices, k-stride 2048 B on both sides.
__device__ __forceinline__ void stage8(float* lds_base, const float* g_base) {
  cp_off<0 * 2048>(lds_base, g_base);
  cp_off<1 * 2048>(lds_base, g_base);
  cp_off<2 * 2048>(lds_base, g_base);
  cp_off<3 * 2048>(lds_base, g_base);
  cp_off<4 * 2048>(lds_base, g_base);
  cp_off<5 * 2048>(lds_base, g_base);
  cp_off<6 * 2048>(lds_base, g_base);
  cp_off<7 * 2048>(lds_base, g_base);
}

// Async loads complete in order: waiting to <=8 outstanding guarantees the
// previous 8-wide batch has landed while the newest batch stays in flight.
__device__ __forceinline__ void wait_async_prev_batch() {
#ifdef HAVE_ASYNC_LDS
  asm volatile("s_wait_asynccnt 8" ::: "memory");
#else
  asm volatile("" ::: "memory");
#endif
}
__device__ __forceinline__ void wait_async_all() {
#ifdef HAVE_ASYNC_LDS
  asm volatile("s_wait_asynccnt 0" ::: "memory");
#else
  asm volatile("" ::: "memory");
#endif
}

// ---------------- Pass 1: per-chunk (A = prod f, B = local scan from 0) ----
__global__ __launch_bounds__(256) void k1_chunk_reduce(
    const float* __restrict__ gi, const float* __restrict__ ge,
    const float* __restrict__ gf, float* __restrict__ Aw,
    float* __restrict__ Bw, int C, int T) {
  __shared__ float fbuf[2 * FBUFSZ];       // 32 KB double buffer
  __shared__ float4 eS[TMAX * K_ / 4];     // 8 KB

  const int tid = threadIdx.x;
  int blk = blockIdx.x;
  const int dt = blk & 1; blk >>= 1;       // D_/DT == 2
  const int c = blk % C;
  const int b = blk / C;
  const int d = dt * DT + tid;
  const int t0 = c * T;

  { // stage e chunk slice [T][K] cooperatively (vectorized)
    const float4* ep = (const float4*)(ge + ((size_t)b * N_ + t0) * K_);
    for (int q = tid; q < T * K_ / 4; q += DT) eS[q] = ep[q];
  }
  __syncthreads();

  const float* fbase = gf + (((size_t)b * N_ + t0) * K_) * (size_t)D_ + d;
  const float* ip = gi + ((size_t)b * N_ + t0) * D_ + d;

  stage8(&fbuf[0] + tid, fbase);           // prologue: stage t=0

  float A[K_], m[K_];
#pragma unroll
  for (int k = 0; k < K_; ++k) { A[k] = 1.0f; m[k] = 0.0f; }

  for (int t = 0; t < T; ++t) {
    const int p = t & 1;
    if (t + 1 < T) {
      stage8(&fbuf[(p ^ 1) * FBUFSZ] + tid, fbase + (size_t)(t + 1) * (K_ * D_));
      wait_async_prev_batch();
    } else {
      wait_async_all();
    }
    const float iv = ip[(size_t)t * D_];
    const float4 e0 = eS[t * 2 + 0];
    const float4 e1 = eS[t * 2 + 1];
    const float ev[K_] = {e0.x, e0.y, e0.z, e0.w, e1.x, e1.y, e1.z, e1.w};
    const float* fb = &fbuf[p * FBUFSZ] + tid;
#pragma unroll
    for (int k = 0; k < K_; ++k) {
      const float fk = fb[k * FSTRIDE];
      A[k] *= fk;
      m[k] = fmaf(fk, m[k], ev[k] * iv);
    }
  }

  float* ap = Aw + (((size_t)b * C + c) * K_) * (size_t)D_ + d;
  float* bp = Bw + (((size_t)b * C + c) * K_) * (size_t)D_ + d;
#pragma unroll
  for (int k = 0; k < K_; ++k) { ap[k * D_] = A[k]; bp[k * D_] = m[k]; }
}

// ---------------- Pass 2: sequential scan over chunk summaries -------------
__global__ __launch_bounds__(256) void k2_chunk_scan(
    const float* __restrict__ Aw, const float* __restrict__ Bw,
    float* __restrict__ Mst, int C) {
  const int idx = blockIdx.x * 256 + threadIdx.x;   // B*K*D threads
  const int d = idx % D_;
  const int k = (idx / D_) % K_;
  const int b = idx / (D_ * K_);
  const size_t kd = (size_t)k * D_ + d;
  float M = 0.0f;
  for (int c = 0; c < C; ++c) {
    const size_t off = (((size_t)b * C + c) * K_) * (size_t)D_ + kd;
    Mst[off] = M;                       // state at chunk start
    M = fmaf(Aw[off], M, Bw[off]);
  }
}

// ---------------- Pass 3: replay chunk from start state, emit outputs ------
__global__ __launch_bounds__(256) void k3_chunk_out(
    const float* __restrict__ gi, const float* __restrict__ ge,
    const float* __restrict__ gf, const float* __restrict__ gs,
    const float* __restrict__ Mst, float* __restrict__ gout, int C, int T) {
  __shared__ float fbuf[2 * FBUFSZ];       // 32 KB
  __shared__ float4 eS[TMAX * K_ / 4];     // 8 KB
  __shared__ float4 sS[TMAX * K_ / 4];     // 8 KB

  const int tid = threadIdx.x;
  int blk = blockIdx.x;
  const int dt = blk & 1; blk >>= 1;
  const int c = blk % C;
  const int b = blk / C;
  const int d = dt * DT + tid;
  const int t0 = c * T;

  {
    const float4* ep = (const float4*)(ge + ((size_t)b * N_ + t0) * K_);
    const float4* sp = (const float4*)(gs + ((size_t)b * N_ + t0) * K_);
    for (int q = tid; q < T * K_ / 4; q += DT) { eS[q] = ep[q]; sS[q] = sp[q]; }
  }
  __syncthreads();

  const float* fbase = gf + (((size_t)b * N_ + t0) * K_) * (size_t)D_ + d;
  const float* ip = gi + ((size_t)b * N_ + t0) * D_ + d;
  float* op = gout + ((size_t)b * N_ + t0) * D_ + d;

  stage8(&fbuf[0] + tid, fbase);

  float m[K_];
  {
    const float* mp = Mst + (((size_t)b * C + c) * K_) * (size_t)D_ + d;
#pragma unroll
    for (int k = 0; k < K_; ++k) m[k] = mp[k * D_];
  }

  for (int t = 0; t < T; ++t) {
    const int p = t & 1;
    if (t + 1 < T) {
      stage8(&fbuf[(p ^ 1) * FBUFSZ] + tid, fbase + (size_t)(t + 1) * (K_ * D_));
      wait_async_prev_batch();
    } else {
      wait_async_all();
    }
    const float iv = ip[(size_t)t * D_];
    const float4 e0 = eS[t * 2 + 0];
    const float4 e1 = eS[t * 2 + 1];
    const float4 s0 = sS[t * 2 + 0];
    const float4 s1 = sS[t * 2 + 1];
    const float ev[K_] = {e0.x, e0.y, e0.z, e0.w, e1.x, e1.y, e1.z, e1.w};
    const float sv[K_] = {s0.x, s0.y, s0.z, s0.w, s1.x, s1.y, s1.z, s1.w};
    const float* fb = &fbuf[p * FBUFSZ] + tid;
    float o = 0.0f;
#pragma unroll
    for (int k = 0; k < K_; ++k) {
      const float fk = fb[k * FSTRIDE];
      m[k] = fmaf(fk, m[k], ev[k] * iv);
      o = fmaf(sv[k], m[k], o);
    }
    __builtin_nontemporal_store(o, op + (size_t)t * D_);
  }
}

extern "C" void kernel_launch(void* const* d_in, const int* in_sizes, int n_in,
                              void* d_out, int out_size, void* d_ws, size_t ws_size,
                              hipStream_t stream) {
  (void)in_sizes; (void)n_in; (void)out_size;
  const float* gi = (const float*)d_in[0];   // i: [b,n,d]
  const float* ge = (const float*)d_in[1];   // e: [b,n,k]
  const float* gf = (const float*)d_in[2];   // f: [b,n,k,d]
  const float* gs = (const float*)d_in[3];   // s: [b,n,k]
  float* out = (float*)d_out;                // o: [b,n,d] fp32

  // Pick chunk count C (power of two, 16..64) to fit workspace: 3 buffers of
  // B*C*K*D floats (A, B, chunk-start states). C=64 -> 12 MB.
  const size_t perC = (size_t)3 * B_ * K_ * D_ * sizeof(float);
  int C = 64;
  while (C > 16 && perC * (size_t)C > ws_size) C >>= 1;
  const int T = N_ / C;   // 64..256 (<= TMAX)

  float* Aw  = (float*)d_ws;
  float* Bw  = Aw + (size_t)B_ * C * K_ * D_;
  float* Mst = Bw + (size_t)B_ * C * K_ * D_;

  const dim3 blk(256);
  const dim3 g1(B_ * C * (D_ / DT));           // 512 blocks at C=64
  k1_chunk_reduce<<<g1, blk, 0, stream>>>(gi, ge, gf, Aw, Bw, C, T);
  k2_chunk_scan<<<dim3((B_ * K_ * D_) / 256), blk, 0, stream>>>(Aw, Bw, Mst, C);
  k3_chunk_out<<<g1, blk, 0, stream>>>(gi, ge, gf, gs, Mst, out, C, T);
}